// V19AlgebraUniversalModelA_ActionZ_67053029425309
// MI455X (gfx1250) — compile-verified
//
#include <hip/hip_runtime.h>
#include <hip/hip_bf16.h>

#define B_   64
#define V_   64
#define N_   16384
#define R_   32
#define Y_   32
#define HID_ 128
#define VP1_ 65
#define BIG_ 1000000000.0f

typedef float v2f __attribute__((ext_vector_type(2)));
typedef float v8f __attribute__((ext_vector_type(8)));

// ---------------------------------------------------------------------------
// Kernel A: build fused mask/sigma byte array  ms[b,n] = matched ? sigma : 0xFF
// plus total0[b] (matched count) and pres0[b] (bitmask of matched sigmas).
// ---------------------------------------------------------------------------
__global__ __launch_bounds__(256) void mask_kernel(
    const int* __restrict__ tables, const int* __restrict__ sigma,
    const int* __restrict__ base_obs, const int* __restrict__ actions,
    const int* __restrict__ responses, const int* __restrict__ t_ptr, int T,
    unsigned char* __restrict__ ms, unsigned int* __restrict__ total0,
    unsigned int* __restrict__ pres0)
{
    const int b = blockIdx.x;
    const int t = t_ptr[0];
    __shared__ unsigned int s_cnt, s_pres;
    if (threadIdx.x == 0) { s_cnt = 0u; s_pres = 0u; }
    __syncthreads();

    const int base = base_obs[b];
    unsigned int lc = 0u, lp = 0u;
    const int* tab_b = tables + (size_t)b * V_ * N_;

    for (int n = threadIdx.x; n < N_; n += blockDim.x) {
        bool m = (tab_b[n] == base);              // row v=0
        for (int i = 0; i < t; ++i) {
            int a = actions[b * T + i];
            if (a != V_) {                        // a == V_ means "stop": keeps m
                int lbl = tab_b[(size_t)a * N_ + n];
                m = m && (lbl == responses[b * T + i]);
            }
        }
        int sg = sigma[(size_t)b * N_ + n];
        ms[(size_t)b * N_ + n] = m ? (unsigned char)sg : (unsigned char)0xFF;
        if (m) { lc++; lp |= (1u << sg); }
    }
    atomicAdd(&s_cnt, lc);
    atomicOr(&s_pres, lp);
    __syncthreads();
    if (threadIdx.x == 0) { total0[b] = s_cnt; pres0[b] = s_pres; }
}

// ---------------------------------------------------------------------------
// Kernel B: one workgroup per (b,v). Streams tables[b,v,:] (the 256 MB hot
// path), LDS-atomic histogram over r in [0,32) with per-r sigma bitmasks.
// Emits raw sums:  sz_raw = sum_r c_r^2 ,  amb_raw = sum_r c_r*popc(pres_r).
// ---------------------------------------------------------------------------
__global__ __launch_bounds__(256) void hist_kernel(
    const int* __restrict__ tables, const unsigned char* __restrict__ ms,
    float* __restrict__ amb_raw, float* __restrict__ sz_raw)
{
    const int bv = blockIdx.x;
    const int b  = bv >> 6;                       // V_ == 64
    __shared__ unsigned int cnt[R_];
    __shared__ unsigned int pres[R_];
    if (threadIdx.x < R_) { cnt[threadIdx.x] = 0u; pres[threadIdx.x] = 0u; }
    __syncthreads();

    const int4*         tp = (const int4*)(tables + (size_t)bv * N_);
    const unsigned int* mp = (const unsigned int*)(ms + (size_t)b * N_);

    for (int i = threadIdx.x; i < N_ / 4; i += blockDim.x) {
        __builtin_prefetch((const void*)(tp + i + 2 * 256), 0, 0);
        int4         tv = tp[i];
        unsigned int mv = mp[i];
        unsigned int m0 =  mv        & 0xFFu;
        unsigned int m1 = (mv >> 8)  & 0xFFu;
        unsigned int m2 = (mv >> 16) & 0xFFu;
        unsigned int m3 = (mv >> 24) & 0xFFu;
        if (m0 != 0xFFu) { atomicAdd(&cnt[tv.x], 1u); atomicOr(&pres[tv.x], 1u << m0); }
        if (m1 != 0xFFu) { atomicAdd(&cnt[tv.y], 1u); atomicOr(&pres[tv.y], 1u << m1); }
        if (m2 != 0xFFu) { atomicAdd(&cnt[tv.z], 1u); atomicOr(&pres[tv.z], 1u << m2); }
        if (m3 != 0xFFu) { atomicAdd(&cnt[tv.w], 1u); atomicOr(&pres[tv.w], 1u << m3); }
    }
    __syncthreads();

    if (threadIdx.x < 32) {
        float c  = (float)cnt[threadIdx.x];
        float u  = (float)__popc(pres[threadIdx.x]);
        float sz = c * c;
        float am = c * u;
        for (int off = 16; off; off >>= 1) {
            sz += __shfl_down(sz, off);
            am += __shfl_down(am, off);
        }
        if (threadIdx.x == 0) { sz_raw[bv] = sz; amb_raw[bv] = am; }
    }
}

// ---------------------------------------------------------------------------
// Kernel C: assemble feat[4160,2] and run the MLP with V_WMMA_F32_16X16X4_F32.
// One wave per 16-row tile (4160 = 260 * 16 exactly -> EXEC always all-ones).
// Layer1: 8 WMMAs over 16-col tiles of HID; exact-erf GeLU; LDS transpose.
// Layer2: K=128 accumulation chain of 32 WMMAs into column 0.
// ---------------------------------------------------------------------------
__global__ __launch_bounds__(32) void mlp_kernel(
    const float* __restrict__ W1, const float* __restrict__ b1,
    const float* __restrict__ W2, const float* __restrict__ b2,
    const int* __restrict__ actions, const int* __restrict__ t_ptr, int T,
    const unsigned int* __restrict__ total0, const unsigned int* __restrict__ pres0,
    const float* __restrict__ amb_raw, const float* __restrict__ sz_raw,
    float* __restrict__ out)
{
    const int lane = threadIdx.x;
    const int row0 = blockIdx.x * 16;
    const int t    = t_ptr[0];
    __shared__ float hbuf[16 * 129];              // padded stride: conflict-free

    // ---- feature assembly for this lane's row (lanes 0..15 own M=0..15) ----
    const int mrow = row0 + (lane & 15);
    const int b    = mrow / VP1_;
    const int vi   = mrow - b * VP1_;
    const unsigned int tot = total0[b];
    const float totf = (float)tot;
    const float safe = fmaxf(totf, 1.0f);
    float amb, szv;
    if (vi == V_) {                               // stop column
        amb = (float)__popc(pres0[b]);
        szv = totf;
    } else {
        amb = amb_raw[b * V_ + vi] / safe;
        szv = sz_raw [b * V_ + vi] / safe;
        bool forb = (vi == 0);
        for (int i = 0; i < t; ++i)
            if (actions[b * T + i] == vi) forb = true;
        if (forb) { amb = BIG_; szv = BIG_; }
    }
    if (tot == 0u) { amb = 0.0f; szv = 0.0f; }

    // A fragment (16x4 f32): lanes 0..15 hold K=0,1 ; lanes 16..31 hold K=2,3 (pad=0)
    v2f afrag;
    afrag.x = (lane < 16) ? amb : 0.0f;
    afrag.y = (lane < 16) ? szv : 0.0f;

    // ---- layer 1: feat @ W1 over 8 column tiles of 16 ----
    const int col = (lane & 15);
    for (int c = 0; c < 8; ++c) {
        const int gcol = col + c * 16;
        v2f bfrag;                                // B (4x16): lanes 0..15 K=0,1
        bfrag.x = (lane < 16) ? W1[gcol]        : 0.0f;
        bfrag.y = (lane < 16) ? W1[HID_ + gcol] : 0.0f;
        v8f d = {};
        d = __builtin_amdgcn_wmma_f32_16x16x4_f32(
                false, afrag, false, bfrag, (short)0, d, false, false);
        // D layout: VGPR j -> (M=j, N=lane) for lanes<16 ; (M=j+8, N=lane-16) else
        const float bb = b1[gcol];
        const int rbase = (lane < 16) ? 0 : 8;
        for (int j = 0; j < 8; ++j) {
            float h = d[j] + bb;
            h = 0.5f * h * (1.0f + erff(h * 0.70710678118654752f));   // exact gelu
            hbuf[(rbase + j) * 129 + gcol] = h;
        }
    }
    __syncthreads();

    // ---- layer 2: h @ W2 (N=1), K=128 in steps of 4, accumulated in WMMA ----
    v8f acc = {};
    for (int k0 = 0; k0 < HID_; k0 += 4) {
        const int m  = lane & 15;
        const int kk = k0 + ((lane < 16) ? 0 : 2);
        v2f a2;
        a2.x = hbuf[m * 129 + kk];
        a2.y = hbuf[m * 129 + kk + 1];
        v2f bcol;                                 // only N==0 column non-zero
        bcol.x = ((lane & 15) == 0) ? W2[kk]     : 0.0f;
        bcol.y = ((lane & 15) == 0) ? W2[kk + 1] : 0.0f;
        acc = __builtin_amdgcn_wmma_f32_16x16x4_f32(
                false, a2, false, bcol, (short)0, acc, false, false);
    }
    const float bias2 = b2[0];
    if ((lane & 15) == 0) {                       // lane 0 -> rows 0..7, lane 16 -> 8..15
        const int rbase = (lane < 16) ? 0 : 8;
        for (int j = 0; j < 8; ++j)
            out[row0 + rbase + j] = acc[j] + bias2;
    }
}

// ---------------------------------------------------------------------------
extern "C" void kernel_launch(void* const* d_in, const int* in_sizes, int n_in,
                              void* d_out, int out_size, void* d_ws, size_t ws_size,
                              hipStream_t stream) {
    const int* tables    = (const int*)d_in[0];
    const int* sigma     = (const int*)d_in[1];
    const int* base_obs  = (const int*)d_in[2];
    const int* actions   = (const int*)d_in[3];
    const int* responses = (const int*)d_in[4];
    const float* W1      = (const float*)d_in[5];
    const float* b1      = (const float*)d_in[6];
    const float* W2      = (const float*)d_in[7];
    const float* b2      = (const float*)d_in[8];
    const int* t_ptr     = (const int*)d_in[9];
    // d_in[10] = max_steps (unused)
    float* out = (float*)d_out;

    const int T = in_sizes[3] / B_;               // actions stride (=8)

    // workspace carve-up
    unsigned char* ms      = (unsigned char*)d_ws;
    unsigned int*  total0  = (unsigned int*)(ms + (size_t)B_ * N_);
    unsigned int*  pres0   = total0 + B_;
    float*         amb_raw = (float*)(pres0 + B_);
    float*         sz_raw  = amb_raw + (size_t)B_ * V_;

    mask_kernel<<<B_, 256, 0, stream>>>(tables, sigma, base_obs, actions,
                                        responses, t_ptr, T, ms, total0, pres0);

    hist_kernel<<<B_ * V_, 256, 0, stream>>>(tables, ms, amb_raw, sz_raw);

    const int ntiles = (B_ * VP1_) / 16;          // 4160 / 16 = 260
    mlp_kernel<<<ntiles, 32, 0, stream>>>(W1, b1, W2, b2, actions, t_ptr, T,
                                          total0, pres0, amb_raw, sz_raw, out);
}